// TVL1OF_54116587929714
// MI455X (gfx1250) — compile-verified
//
#include <hip/hip_runtime.h>
#include <cstdint>

#define HH 512
#define WW 512
#define NPIX (HH * WW)          // 262144 = 2^18
#define BATCH 4
#define NB (BATCH * NPIX)       // 1048576 floats per field
#define N_ITER 20
#define EPSF 1e-8f

typedef __attribute__((ext_vector_type(2))) float v2f;
typedef __attribute__((ext_vector_type(8))) float v8f;

// ---------- CDNA5 async copy: global -> LDS (tracked by ASYNCcnt) ----------
__device__ __forceinline__ void async_ld_f32(uint32_t lds_byte_off,
                                             const float* base,
                                             uint32_t glob_byte_off) {
  // vdst = LDS byte address VGPR, vaddr = 32-bit offset VGPR, saddr = 64-bit base SGPRs
  asm volatile("global_load_async_to_lds_b32 %0, %1, %2"
               :
               : "v"(lds_byte_off), "v"(glob_byte_off), "s"(base)
               : "memory");
}
__device__ __forceinline__ void wait_async0() {
  asm volatile("s_wait_asynccnt 0" ::: "memory");
}

// =======================================================================
// Setup: rho_c, image gradients, norms; zero-init u and p.
// =======================================================================
__global__ __launch_bounds__(256) void k_setup(
    const float* __restrict__ x, float* rho_c, float* gx, float* gy,
    float* nrm, float* inv, float* u1, float* u2,
    float* p11, float* p12, float* p21, float* p22) {
  int gid = blockIdx.x * 256 + threadIdx.x;
  if (gid >= NB) return;
  int b = gid >> 18;
  int pix = gid & (NPIX - 1);
  int r = pix >> 9, c = pix & 511;
  const float* x0 = x + (size_t)b * 2 * NPIX;
  const float* x1 = x0 + NPIX;
  auto X = [&](int rr, int cc) -> float {
    return (rr >= 0 && rr < HH && cc >= 0 && cc < WW) ? x1[rr * WW + cc] : 0.f;
  };
  float gxx = (1.f / 6.f) * (-X(r-1,c-1) + X(r-1,c+1)
                             - 2.f*X(r,c-1) + 2.f*X(r,c+1)
                             - X(r+1,c-1) + X(r+1,c+1));
  float gyy = (1.f / 6.f) * (-X(r-1,c-1) - 2.f*X(r-1,c) - X(r-1,c+1)
                             + X(r+1,c-1) + 2.f*X(r+1,c) + X(r+1,c+1));
  float nn = gxx * gxx + gyy * gyy + EPSF;
  rho_c[gid] = x1[pix] - x0[pix];
  gx[gid] = gxx; gy[gid] = gyy; nrm[gid] = nn; inv[gid] = 1.f / nn;
  u1[gid] = 0.f; u2[gid] = 0.f;
  p11[gid] = 0.f; p12[gid] = 0.f; p21[gid] = 0.f; p22[gid] = 0.f;
}

// =======================================================================
// u-update: v = u - shrink(rho);  u_new = v + theta * div(p)
// p staged to LDS (with halo) via async-to-LDS.
// =======================================================================
__global__ __launch_bounds__(256) void k_u(
    const float* __restrict__ u1in, const float* __restrict__ u2in,
    const float* __restrict__ p11, const float* __restrict__ p12,
    const float* __restrict__ p21, const float* __restrict__ p22,
    const float* __restrict__ rho_c, const float* __restrict__ gx,
    const float* __restrict__ gy, const float* __restrict__ nrm,
    const float* __restrict__ inv,
    float* __restrict__ u1out, float* __restrict__ u2out,
    const float* __restrict__ lamp, const float* __restrict__ thetap,
    const float* __restrict__ wxp, const float* __restrict__ wyp) {
  __shared__ float sp11[32][34];  // rows 0..31, cols -1..32
  __shared__ float sp12[34][32];  // rows -1..32, cols 0..31
  __shared__ float sp21[32][34];
  __shared__ float sp22[34][32];

  int tid = threadIdx.x;
  int gb = blockIdx.x;
  int b = gb >> 8;           // 256 tiles per batch sample
  int t = gb & 255;
  int tr = (t >> 4) << 5;    // 16x16 grid of 32x32 tiles
  int tc = (t & 15) << 5;

  const float* P11 = p11 + (size_t)b * NPIX;
  const float* P12 = p12 + (size_t)b * NPIX;
  const float* P21 = p21 + (size_t)b * NPIX;
  const float* P22 = p22 + (size_t)b * NPIX;

  uint32_t l11 = (uint32_t)(uintptr_t)&sp11[0][0];
  uint32_t l12 = (uint32_t)(uintptr_t)&sp12[0][0];
  uint32_t l21 = (uint32_t)(uintptr_t)&sp21[0][0];
  uint32_t l22 = (uint32_t)(uintptr_t)&sp22[0][0];

  // x-halo tiles (clamped addresses; zero-padding applied at use)
  for (int e = tid; e < 32 * 34; e += 256) {
    int i = e / 34, jj = e % 34;
    int gj = tc + jj - 1;
    gj = gj < 0 ? 0 : (gj >= WW ? WW - 1 : gj);
    uint32_t off = (uint32_t)(((tr + i) * WW + gj) * 4);
    async_ld_f32(l11 + e * 4, P11, off);
    async_ld_f32(l21 + e * 4, P21, off);
  }
  // y-halo tiles
  for (int e = tid; e < 34 * 32; e += 256) {
    int i = e / 32, j = e % 32;
    int gi = tr + i - 1;
    gi = gi < 0 ? 0 : (gi >= HH ? HH - 1 : gi);
    uint32_t off = (uint32_t)((gi * WW + tc + j) * 4);
    async_ld_f32(l12 + e * 4, P12, off);
    async_ld_f32(l22 + e * 4, P22, off);
  }
  wait_async0();
  __syncthreads();

  float theta = thetap[0];
  float tl = theta * lamp[0];
  float wx0 = wxp[0], wx1 = wxp[1], wx2 = wxp[2];
  float wy0 = wyp[0], wy1 = wyp[1], wy2 = wyp[2];

  for (int e = tid; e < 1024; e += 256) {
    int i = e >> 5, j = e & 31;
    int gi = tr + i, gj = tc + j;
    int idx = b * NPIX + gi * WW + gj;
    float U1 = u1in[idx], U2 = u2in[idx];
    float GX = gx[idx], GY = gy[idx];
    float rho = rho_c[idx] + GX * U1 + GY * U2;
    float th = tl * nrm[idx];
    float sgn = (rho > 0.f) ? 1.f : ((rho < 0.f) ? -1.f : 0.f);
    float d = (fabsf(rho) < th) ? rho * inv[idx] : tl * sgn;
    float v1 = U1 - d * GX;
    float v2 = U2 - d * GY;

    float a0 = (gj >= 1)     ? sp11[i][j]     : 0.f;
    float a2 = (gj + 1 < WW) ? sp11[i][j + 2] : 0.f;
    float b0 = (gi >= 1)     ? sp12[i][j]     : 0.f;
    float b2 = (gi + 1 < HH) ? sp12[i + 2][j] : 0.f;
    float div1 = wx0 * a0 + wx1 * sp11[i][j + 1] + wx2 * a2
               + wy0 * b0 + wy1 * sp12[i + 1][j] + wy2 * b2;

    float c0 = (gj >= 1)     ? sp21[i][j]     : 0.f;
    float c2 = (gj + 1 < WW) ? sp21[i][j + 2] : 0.f;
    float e0 = (gi >= 1)     ? sp22[i][j]     : 0.f;
    float e2 = (gi + 1 < HH) ? sp22[i + 2][j] : 0.f;
    float div2 = wx0 * c0 + wx1 * sp21[i][j + 1] + wx2 * c2
               + wy0 * e0 + wy1 * sp22[i + 1][j] + wy2 * e2;

    u1out[idx] = v1 + theta * div1;
    u2out[idx] = v2 + theta * div2;
  }
}

// =======================================================================
// p-update (pointwise in p, in-place): p = (p + r*grad(u)) / (1 + r*|grad(u)|_1)
// new u staged to LDS (right/bottom halo) via async-to-LDS.
// =======================================================================
__global__ __launch_bounds__(256) void k_p(
    const float* __restrict__ u1, const float* __restrict__ u2,
    float* __restrict__ p11, float* __restrict__ p12,
    float* __restrict__ p21, float* __restrict__ p22,
    const float* __restrict__ taup, const float* __restrict__ thetap) {
  __shared__ float su1[33][34];
  __shared__ float su2[33][34];

  int tid = threadIdx.x;
  int gb = blockIdx.x;
  int b = gb >> 8;
  int t = gb & 255;
  int tr = (t >> 4) << 5;
  int tc = (t & 15) << 5;

  const float* U1 = u1 + (size_t)b * NPIX;
  const float* U2 = u2 + (size_t)b * NPIX;
  uint32_t l1 = (uint32_t)(uintptr_t)&su1[0][0];
  uint32_t l2 = (uint32_t)(uintptr_t)&su2[0][0];

  for (int e = tid; e < 33 * 33; e += 256) {
    int i = e / 33, j = e % 33;
    int gi = tr + i; if (gi >= HH) gi = HH - 1;
    int gj = tc + j; if (gj >= WW) gj = WW - 1;
    uint32_t off = (uint32_t)((gi * WW + gj) * 4);
    uint32_t lo = (uint32_t)((i * 34 + j) * 4);
    async_ld_f32(l1 + lo, U1, off);
    async_ld_f32(l2 + lo, U2, off);
  }
  wait_async0();
  __syncthreads();

  float r = taup[0] / thetap[0];
  for (int e = tid; e < 1024; e += 256) {
    int i = e >> 5, j = e & 31;
    int gi = tr + i, gj = tc + j;
    int idx = b * NPIX + gi * WW + gj;

    float c1 = su1[i][j];
    float r1 = (gj + 1 < WW) ? su1[i][j + 1] : 0.f;
    float d1 = (gi + 1 < HH) ? su1[i + 1][j] : 0.f;
    float g1x = r1 - c1, g1y = d1 - c1;
    float den1 = 1.f + r * (fabsf(g1x) + fabsf(g1y));
    p11[idx] = (p11[idx] + r * g1x) / den1;
    p12[idx] = (p12[idx] + r * g1y) / den1;

    float c2 = su2[i][j];
    float r2 = (gj + 1 < WW) ? su2[i][j + 1] : 0.f;
    float d2 = (gi + 1 < HH) ? su2[i + 1][j] : 0.f;
    float g2x = r2 - c2, g2y = d2 - c2;
    float den2 = 1.f + r * (fabsf(g2x) + fabsf(g2y));
    p21[idx] = (p21[idx] + r * g2x) / den2;
    p22[idx] = (p22[idx] + r * g2y) / den2;
  }
}

// =======================================================================
// AvgPool 3x3 (stride 1, pad 1, count_include_pad):
// column pass with VALU into LDS, row pass as a banded 16x18 x 18x16 GEMM
// via five chained v_wmma_f32_16x16x4_f32 (K zero-padded to 20 in LDS so all
// fragment loads are unconditional / non-divergent).
// One wave per 16x16 output tile; 8 waves per block.
// =======================================================================
__global__ __launch_bounds__(256) void k_avgpool(
    const float* __restrict__ u1, const float* __restrict__ u2,
    float* __restrict__ out) {
  // Per-wave column-sum tile: 16 rows x 20 K-cols (cols 0..17 = u cols -1..16,
  // cols 18..19 = zero pad). Stride 20 floats -> 32 lanes' fragment reads hit
  // 32 distinct banks (m*20 mod 64 distinct for m=0..15; halves offset by +2).
  __shared__ float S[8][16][20];

  int lane = threadIdx.x & 31;
  int w = threadIdx.x >> 5;
  int tile = blockIdx.x * 8 + w;
  int plane = tile >> 10;          // 1024 tiles per 512x512 plane
  int t = tile & 1023;
  int tr = (t >> 5) << 4;
  int tc = (t & 31) << 4;
  int b = plane >> 1, ch = plane & 1;
  const float* U = (ch ? u2 : u1) + (size_t)b * NPIX;

  // Column 3-tap sums over 18 columns + zero pad (zero padding at image edges).
  for (int e = lane; e < 16 * 20; e += 32) {
    int i = e / 20, jc = e % 20;
    int col = tc + jc - 1;
    float s = 0.f;
    if (jc < 18 && col >= 0 && col < WW) {
      int row = tr + i;
      if (row - 1 >= 0) s += U[(row - 1) * WW + col];
      s += U[row * WW + col];
      if (row + 1 < HH) s += U[(row + 1) * WW + col];
    }
    S[w][i][jc] = s;
  }
  // Wave-private LDS region; DS ops are in-order within a wave.

  // Row pass: out(M,N) = sum_K S(M,K) * T(K,N), T(K,N) = 1/9 iff K-1-N in [-1,1]
  // (rows K=18,19 of T are all-zero by the band test since N <= 15).
  // f32 WMMA fragment layouts: A/B: K = kbase + 2*laneHalf + vgprIdx;
  // A value index M = lane%16, B value index N = lane%16.
  int half = lane >> 4;
  int m = lane & 15;
  v8f acc = {};
  for (int kc = 0; kc < 5; ++kc) {
    int kb = kc * 4;
    v2f a, bm;
    for (int j = 0; j < 2; ++j) {
      int K = kb + 2 * half + j;
      a[j] = S[w][m][K];
      int dn = (K - 1) - m;
      bm[j] = (dn >= -1 && dn <= 1) ? (1.f / 9.f) : 0.f;
    }
    acc = __builtin_amdgcn_wmma_f32_16x16x4_f32(
        false, a, false, bm, (short)0, acc, false, false);
  }

  // C/D layout: VGPR v holds row M = v + 8*laneHalf, col N = lane%16.
  float* O = out + (size_t)plane * NPIX;
  for (int v = 0; v < 8; ++v) {
    int M = v + 8 * half;
    O[(tr + M) * WW + tc + m] = acc[v];
  }
}

// =======================================================================
extern "C" void kernel_launch(void* const* d_in, const int* in_sizes, int n_in,
                              void* d_out, int out_size, void* d_ws, size_t ws_size,
                              hipStream_t stream) {
  const float* x     = (const float*)d_in[0];
  const float* lam   = (const float*)d_in[1];
  const float* tau   = (const float*)d_in[2];
  const float* theta = (const float*)d_in[3];
  const float* wx    = (const float*)d_in[4];
  const float* wy    = (const float*)d_in[5];

  float* ws = (float*)d_ws;
  float* rho_c = ws + 0 * (size_t)NB;
  float* gx    = ws + 1 * (size_t)NB;
  float* gy    = ws + 2 * (size_t)NB;
  float* nrm   = ws + 3 * (size_t)NB;
  float* inv   = ws + 4 * (size_t)NB;
  float* uA1   = ws + 5 * (size_t)NB;
  float* uA2   = ws + 6 * (size_t)NB;
  float* uB1   = ws + 7 * (size_t)NB;
  float* uB2   = ws + 8 * (size_t)NB;
  float* p11   = ws + 9 * (size_t)NB;
  float* p12   = ws + 10 * (size_t)NB;
  float* p21   = ws + 11 * (size_t)NB;
  float* p22   = ws + 12 * (size_t)NB;

  k_setup<<<NB / 256, 256, 0, stream>>>(x, rho_c, gx, gy, nrm, inv,
                                        uA1, uA2, p11, p12, p21, p22);

  float* cur1 = uA1; float* cur2 = uA2;
  float* nxt1 = uB1; float* nxt2 = uB2;
  for (int it = 0; it < N_ITER; ++it) {
    k_u<<<1024, 256, 0, stream>>>(cur1, cur2, p11, p12, p21, p22,
                                  rho_c, gx, gy, nrm, inv,
                                  nxt1, nxt2, lam, theta, wx, wy);
    k_p<<<1024, 256, 0, stream>>>(nxt1, nxt2, p11, p12, p21, p22, tau, theta);
    float* s1 = cur1; cur1 = nxt1; nxt1 = s1;
    float* s2 = cur2; cur2 = nxt2; nxt2 = s2;
  }

  k_avgpool<<<1024, 256, 0, stream>>>(cur1, cur2, (float*)d_out);
}